// R_transformations_919123002022
// MI455X (gfx1250) — compile-verified
//
#include <hip/hip_runtime.h>
#include <math.h>

typedef float v2f __attribute__((ext_vector_type(2)));
typedef float v8f __attribute__((ext_vector_type(8)));
typedef long long i64;

#define B_   8
#define S1_  256
#define M2_  129
#define XY0_ (S1_ * M2_)   // 33024

__device__ __forceinline__ float gelu_t(float x) {
  float x3 = x * x * x;
  return 0.5f * x * (1.0f + tanhf(0.7978845608028654f * (x + 0.044715f * x3)));
}

// ---------------------------------------------------------------------------
// Stage 1: k/Re feature build + complex p-MLP (9->32 cgelu, 32->32),
// x *= re_feature, sparsity mask, +1  -> planar (B,32,S1,M2) real/imag.
// Elementwise/transcendental dominated -> VALU kernel, weights in LDS.
// ---------------------------------------------------------------------------
__global__ void pmlp_kernel(const float* __restrict__ xr, const float* __restrict__ xi,
                            const float* __restrict__ Re,
                            const float* __restrict__ w1r, const float* __restrict__ w1i,
                            const float* __restrict__ b1r, const float* __restrict__ b1i,
                            const float* __restrict__ w2r, const float* __restrict__ w2i,
                            const float* __restrict__ b2r, const float* __restrict__ b2i,
                            float* __restrict__ outR, float* __restrict__ outI) {
  __shared__ float sW1r[288], sW1i[288], sW2r[1024], sW2i[1024];
  __shared__ float sb1r[32], sb1i[32], sb2r[32], sb2i[32];
  int tid = threadIdx.x;
  for (int i = tid; i < 288; i += blockDim.x) { sW1r[i] = w1r[i]; sW1i[i] = w1i[i]; }
  for (int i = tid; i < 1024; i += blockDim.x) { sW2r[i] = w2r[i]; sW2i[i] = w2i[i]; }
  if (tid < 32) {
    sb1r[tid] = b1r[tid]; sb1i[tid] = b1i[tid];
    sb2r[tid] = b2r[tid]; sb2i[tid] = b2i[tid];
  }
  __syncthreads();

  int idx = blockIdx.x * blockDim.x + tid;
  if (idx >= B_ * XY0_) return;
  int b = idx / XY0_, pt = idx % XY0_;
  int x = pt / M2_, y = pt % M2_;
  float kx = (x < 128) ? (float)x : (float)(x - 256);
  float ky = (y < 128) ? (float)y : (float)(y - 256);

  // feat order: kx^(1/3), ky^(1/3), kx^(2/3), ky^(2/3), kx, ky, Re^(1/3), Re^(2/3), Re
  float fr[9], fi[9];
  const float c13 = 0.5f, s13 = 0.8660254037844386f;
  const float c23 = -0.5f, s23 = 0.8660254037844386f;
  {
    float a = fabsf(kx); float r1 = powf(a, 0.33333333333333333f); float r2 = r1 * r1;
    if (kx < 0.f) { fr[0] = r1 * c13; fi[0] = r1 * s13; fr[2] = r2 * c23; fi[2] = r2 * s23; }
    else          { fr[0] = r1;       fi[0] = 0.f;      fr[2] = r2;       fi[2] = 0.f; }
    fr[4] = kx; fi[4] = 0.f;
  }
  {
    float a = fabsf(ky); float r1 = powf(a, 0.33333333333333333f); float r2 = r1 * r1;
    if (ky < 0.f) { fr[1] = r1 * c13; fi[1] = r1 * s13; fr[3] = r2 * c23; fi[3] = r2 * s23; }
    else          { fr[1] = r1;       fi[1] = 0.f;      fr[3] = r2;       fi[3] = 0.f; }
    fr[5] = ky; fi[5] = 0.f;
  }
  float rev = Re[b];
  fr[6] = powf(rev, 0.33333333333333333f); fi[6] = 0.f;
  fr[7] = fr[6] * fr[6];                   fi[7] = 0.f;
  fr[8] = rev;                             fi[8] = 0.f;

  // layer 1: h[o] = sum_i feat[i]*W1[o,i] + b1[o]; cgelu
  float hr[32], hi[32];
  for (int o = 0; o < 32; ++o) {
    float ar = sb1r[o], ai = sb1i[o];
    for (int i = 0; i < 9; ++i) {
      float wr = sW1r[o * 9 + i], wi = sW1i[o * 9 + i];
      ar += fr[i] * wr - fi[i] * wi;
      ai += fr[i] * wi + fi[i] * wr;
    }
    hr[o] = gelu_t(ar); hi[o] = gelu_t(ai);
  }
  // layer 2: rf[o] = sum_i h[i]*W2[o,i] + b2[o]
  float rfr[32], rfi[32];
  for (int o = 0; o < 32; ++o) {
    float ar = sb2r[o], ai = sb2i[o];
    for (int i = 0; i < 32; ++i) {
      float wr = sW2r[o * 32 + i], wi = sW2i[o * 32 + i];
      ar += hr[i] * wr - hi[i] * wi;
      ai += hr[i] * wi + hi[i] * wr;
    }
    rfr[o] = ar; rfi[o] = ai;
  }
  // x *= rf ; mask |z|>0.05 ; +1 (real only)
  for (int c = 0; c < 32; ++c) {
    i64 off = ((i64)(b * 32 + c)) * XY0_ + pt;
    float axr = xr[off], axi = xi[off];
    float zr = axr * rfr[c] - axi * rfi[c];
    float zi = axr * rfi[c] + axi * rfr[c];
    float m2 = zr * zr + zi * zi;
    if (m2 <= 0.05f * 0.05f) { zr = 0.f; zi = 0.f; }
    outR[off] = zr + 1.0f;
    outI[off] = zi;
  }
}

// ---------------------------------------------------------------------------
// Generic complex GEMM via V_WMMA_F32_16X16X4_F32:
//   Out(Cout x P) = W^T (Cout x Cin) * In(Cin x P), per batch.
// Each wave computes TWO 16x16 output tiles (32 output channels) for one
// 16-point tile, so every B-fragment (the streamed input) feeds 8 WMMAs —
// halves/quarters input load traffic vs one-tile-per-wave.
// 4 fp32 accumulators per tile for the complex product; epilogue fuses
// cgelu, optional +1, and optional corner scatter-add (up path). Input
// gather (down-path corner truncation) is fused into B-fragment loads.
// ---------------------------------------------------------------------------
__global__ void cgemm_wmma(const float* __restrict__ Wr, const float* __restrict__ Wi,
                           const float* __restrict__ Inr, const float* __restrict__ Ini,
                           float* __restrict__ Outr, float* __restrict__ Outi,
                           int Cin, int Cout,
                           int Xsrc, int Ysrc,      // source geometry
                           int Xl, int Yl,          // logical GEMM domain (P = Xl*Yl)
                           int identIn,             // 1 -> linear source indexing
                           int addOne,
                           int scatter,             // 1 -> corner scatter-add into Out
                           int Xprev, int Yprev) {  // dst geometry when scatter
  const int P = Xl * Yl;
  const int nPtT = P >> 4;
  const int nCoP = Cout >> 5;            // pairs of 16-wide Cout tiles
  const i64 totalWaves = (i64)B_ * nCoP * nPtT;

  int lane = threadIdx.x & 31;
  i64 gw = (i64)blockIdx.x * (blockDim.x >> 5) + (threadIdx.x >> 5);
  if (gw >= totalWaves) return;  // wave-uniform: EXEC stays all-ones for WMMA

  int ptT = (int)(gw % nPtT);
  i64 t = gw / nPtT;
  int coP = (int)(t % nCoP);
  int b = (int)(t / nCoP);
  int pt0 = ptT << 4, co0 = coP << 5;    // this wave: Cout rows [co0, co0+32)
  int n = lane & 15, half = lane >> 4;

  // source offset of this lane's point (same point for both lane halves)
  int p = pt0 + n;
  i64 poff;
  if (identIn) {
    poff = p;
  } else {
    int xl = p / Yl, yl = p % Yl;
    int sx = (xl < (Xl >> 1)) ? xl : (xl + (Xsrc - Xl));
    poff = (i64)sx * Ysrc + yl;
  }
  const i64 XYs = (i64)Xsrc * Ysrc;
  const i64 inB = (i64)b * Cin * XYs;

  v8f aRR[2], aII[2], aRI[2], aIR[2];
#pragma unroll
  for (int tdx = 0; tdx < 2; ++tdx) {
    aRR[tdx] = (v8f){0.f,0.f,0.f,0.f,0.f,0.f,0.f,0.f};
    aII[tdx] = (v8f){0.f,0.f,0.f,0.f,0.f,0.f,0.f,0.f};
    aRI[tdx] = (v8f){0.f,0.f,0.f,0.f,0.f,0.f,0.f,0.f};
    aIR[tdx] = (v8f){0.f,0.f,0.f,0.f,0.f,0.f,0.f,0.f};
  }

  for (int k0 = 0; k0 < Cin; k0 += 4) {
    int ka = k0 + 2 * half;        // K rows this half-wave supplies
    int kb = ka + 1;
    // B fragment: In[K][pt0+n] — 16 contiguous floats per half-wave per VGPR
    v2f br, bi;
    br.x = Inr[inB + (i64)ka * XYs + poff]; br.y = Inr[inB + (i64)kb * XYs + poff];
    bi.x = Ini[inB + (i64)ka * XYs + poff]; bi.y = Ini[inB + (i64)kb * XYs + poff];
#pragma unroll
    for (int tdx = 0; tdx < 2; ++tdx) {
      int wm = co0 + tdx * 16 + n;
      // A fragment: A[M=wm][K] = W[K][wm]  (weights, cache-resident)
      v2f ar, ai;
      ar.x = Wr[ka * Cout + wm]; ar.y = Wr[kb * Cout + wm];
      ai.x = Wi[ka * Cout + wm]; ai.y = Wi[kb * Cout + wm];
      aRR[tdx] = __builtin_amdgcn_wmma_f32_16x16x4_f32(false, ar, false, br, (short)0, aRR[tdx], false, false);
      aII[tdx] = __builtin_amdgcn_wmma_f32_16x16x4_f32(false, ai, false, bi, (short)0, aII[tdx], false, false);
      aRI[tdx] = __builtin_amdgcn_wmma_f32_16x16x4_f32(false, ar, false, bi, (short)0, aRI[tdx], false, false);
      aIR[tdx] = __builtin_amdgcn_wmma_f32_16x16x4_f32(false, ai, false, br, (short)0, aIR[tdx], false, false);
    }
  }

  // Epilogue: complex combine, cgelu, optional +1, planar store or scatter-add
  const i64 XYl = (i64)Xl * Yl;
  const i64 XYp = (i64)Xprev * Yprev;
#pragma unroll
  for (int tdx = 0; tdx < 2; ++tdx) {
    for (int j = 0; j < 8; ++j) {
      float re = aRR[tdx][j] - aII[tdx][j];
      float im = aRI[tdx][j] + aIR[tdx][j];
      re = gelu_t(re);
      im = gelu_t(im);
      if (addOne) re += 1.0f;
      int o = co0 + tdx * 16 + j + 8 * half;  // D layout: VGPR j -> rows j / j+8
      if (!scatter) {
        i64 off = ((i64)b * Cout + o) * XYl + p;
        Outr[off] = re; Outi[off] = im;
      } else {
        int xl = p / Yl, yl = p % Yl;
        int m1 = Xl >> 1;
        int dx = (xl < m1) ? xl : (xl + (Xprev - Xl));
        i64 off = ((i64)b * Cout + o) * XYp + (i64)dx * Yprev + yl;
        Outr[off] += re; Outi[off] += im;   // unique writer per element
      }
    }
  }
}

// planar real/imag -> interleaved complex64 in d_out
__global__ void interleave_kernel(const float* __restrict__ r, const float* __restrict__ im,
                                  float* __restrict__ dst, i64 n) {
  i64 idx = (i64)blockIdx.x * blockDim.x + threadIdx.x;
  if (idx >= n) return;
  dst[2 * idx]     = r[idx];
  dst[2 * idx + 1] = im[idx];
}

extern "C" void kernel_launch(void* const* d_in, const int* in_sizes, int n_in,
                              void* d_out, int out_size, void* d_ws, size_t ws_size,
                              hipStream_t stream) {
  const float* xr   = (const float*)d_in[0];
  const float* xi   = (const float*)d_in[1];
  const float* Re   = (const float*)d_in[2];
  const float* pw1r = (const float*)d_in[3];
  const float* pw1i = (const float*)d_in[4];
  const float* pb1r = (const float*)d_in[5];
  const float* pb1i = (const float*)d_in[6];
  const float* pw2r = (const float*)d_in[7];
  const float* pw2i = (const float*)d_in[8];
  const float* pb2r = (const float*)d_in[9];
  const float* pb2i = (const float*)d_in[10];
  const float* w0r  = (const float*)d_in[11];
  const float* w0i  = (const float*)d_in[12];
  const float* wd1r = (const float*)d_in[13];
  const float* wd1i = (const float*)d_in[14];
  const float* wu1r = (const float*)d_in[15];
  const float* wu1i = (const float*)d_in[16];
  const float* wd2r = (const float*)d_in[17];
  const float* wd2i = (const float*)d_in[18];
  const float* wu2r = (const float*)d_in[19];
  const float* wu2i = (const float*)d_in[20];
  // d_in[21]=S1, d_in[22]=S2 fixed at 256 (compile-time constants here)

  const i64 NP0 = (i64)B_ * 32 * XY0_;       // 8,454,144  (B,32,256,129)
  const i64 NP1 = (i64)B_ * 64 * 128 * 64;   // 4,194,304  (B,64,128,64)
  const i64 NP2 = (i64)B_ * 64 * 64 * 32;    // 1,048,576  (B,64,64,32)

  float* ws   = (float*)d_ws;
  float* xinR = ws;             float* xinI = xinR + NP0;
  float* xs0R = xinI + NP0;     float* xs0I = xs0R + NP0;
  float* xs1R = xs0I + NP0;     float* xs1I = xs1R + NP1;
  float* xs2R = xs1I + NP1;     float* xs2I = xs2R + NP2;

  // 1) features + p-MLP + mask + (+1)
  {
    int total = B_ * XY0_;
    pmlp_kernel<<<(total + 255) / 256, 256, 0, stream>>>(
        xr, xi, Re, pw1r, pw1i, pb1r, pb1i, pw2r, pw2i, pb2r, pb2i, xinR, xinI);
  }

  auto launch_gemm = [&](const float* Wr, const float* Wi,
                         const float* Ir, const float* Ii,
                         float* Or, float* Oi,
                         int Cin, int Cout, int Xsrc, int Ysrc,
                         int Xl, int Yl, int identIn, int addOne,
                         int scatter, int Xprev, int Yprev) {
    i64 P = (i64)Xl * Yl;
    i64 totalWaves = (i64)B_ * (Cout / 32) * (P / 16);
    i64 blocks = (totalWaves + 7) / 8;   // 8 waves (256 threads) per block
    cgemm_wmma<<<dim3((unsigned)blocks), 256, 0, stream>>>(
        Wr, Wi, Ir, Ii, Or, Oi, Cin, Cout, Xsrc, Ysrc, Xl, Yl,
        identIn, addOne, scatter, Xprev, Yprev);
  };

  // 2) level 0:  xs0 = cgelu((xin) @ w0)
  launch_gemm(w0r, w0i, xinR, xinI, xs0R, xs0I, 32, 32, 256, 129, 256, 129, 1, 0, 0, 0, 0);
  // 3) down 1:  xs1 = cgelu(corner(xs0) @ wd1) + 1      (corners 64x64 of 256x129)
  launch_gemm(wd1r, wd1i, xs0R, xs0I, xs1R, xs1I, 32, 64, 256, 129, 128, 64, 0, 1, 0, 0, 0);
  // 4) down 2:  xs2 = cgelu(corner(xs1) @ wd2) + 1      (corners 32x32 of 128x64)
  launch_gemm(wd2r, wd2i, xs1R, xs1I, xs2R, xs2I, 64, 64, 128, 64, 64, 32, 0, 1, 0, 0, 0);
  // 5) up 2:    xs1 += corner-scatter(cgelu(xs2 @ wu2))
  launch_gemm(wu2r, wu2i, xs2R, xs2I, xs1R, xs1I, 64, 64, 64, 32, 64, 32, 1, 0, 1, 128, 64);
  // 6) up 1:    xs0 += corner-scatter(cgelu(xs1 @ wu1))
  launch_gemm(wu1r, wu1i, xs1R, xs1I, xs0R, xs0I, 64, 32, 128, 64, 128, 64, 1, 0, 1, 256, 129);

  // 7) pack outputs as interleaved complex64: (xs0, xs0, xs1, xs2)
  float* dout = (float*)d_out;
  {
    i64 n = NP0;
    interleave_kernel<<<(unsigned)((n + 255) / 256), 256, 0, stream>>>(xs0R, xs0I, dout, n);
    interleave_kernel<<<(unsigned)((n + 255) / 256), 256, 0, stream>>>(xs0R, xs0I, dout + 2 * NP0, n);
  }
  {
    i64 n = NP1;
    interleave_kernel<<<(unsigned)((n + 255) / 256), 256, 0, stream>>>(xs1R, xs1I, dout + 4 * NP0, n);
  }
  {
    i64 n = NP2;
    interleave_kernel<<<(unsigned)((n + 255) / 256), 256, 0, stream>>>(xs2R, xs2I, dout + 4 * NP0 + 2 * NP1, n);
  }
}